// DecoderLSTM_64141041598671
// MI455X (gfx1250) — compile-verified
//
#include <hip/hip_runtime.h>

// Problem sizes (fixed by the reference)
#define B_   128
#define T_   20
#define E_   300
#define H_   512
#define V_   10000
#define G4H_ (4 * H_)   // 2048

typedef __attribute__((ext_vector_type(2))) float v2f;
typedef __attribute__((ext_vector_type(8))) float v8f;

// D = A(16x4 f32) * B(4x16 f32) + C(16x16 f32)  -- native CDNA5 fp32 WMMA
__device__ __forceinline__ v8f wmma_f32(v2f a, v2f b, v8f c) {
    return __builtin_amdgcn_wmma_f32_16x16x4_f32(
        /*neg_a=*/false, a, /*neg_b=*/false, b,
        /*c_mod=*/(short)0, c, /*reuse_a=*/false, /*reuse_b=*/false);
}

__device__ __forceinline__ float sigmoidf_(float x) {
    return 1.0f / (1.0f + __expf(-x));
}

// ---------------------------------------------------------------------------
// Kernel 1: xg[t*B+b][g] = sum_e embed[captions[b,t]][e] * W_ih[g][e] + b_ih[g] + b_hh[g]
// GEMM M=T*B=2560, N=4H=2048, K=E=300.  grid=(160,32), block=128 (4 waves).
// ---------------------------------------------------------------------------
__global__ __launch_bounds__(128) void xg_kernel(
    const int*   __restrict__ captions,   // [B,T]
    const float* __restrict__ embed,      // [V,E]
    const float* __restrict__ W_ih,       // [4H,E]
    const float* __restrict__ b_ih,       // [4H]
    const float* __restrict__ b_hh,       // [4H]
    float*       __restrict__ xg)         // [T,B,4H]
{
    const int lane = threadIdx.x & 31;
    const int wave = threadIdx.x >> 5;
    const int col  = lane & 15;             // M or N index within tile
    const int kb   = (lane >> 4) << 1;      // K sub-offset: 0 or 2
    const int m0   = blockIdx.x * 16;       // row tile base (r = t*B + b)
    const int n0   = blockIdx.y * 64 + wave * 16;

    // A row: embedding of token for row r = m0+col
    const int r   = m0 + col;
    const int t   = r / B_;
    const int b   = r - t * B_;
    const int tok = captions[b * T_ + t];
    const float* arow = embed + (size_t)tok * E_;
    const float* brow = W_ih + (size_t)(n0 + col) * E_;   // B[k][n] = W_ih[n][k]

    v8f acc = {};
    for (int k = 0; k < E_; k += 4) {       // 300/4 = 75 iterations, exact
        v2f a, bb;
        a.x  = arow[k + kb];  a.y  = arow[k + kb + 1];
        bb.x = brow[k + kb];  bb.y = brow[k + kb + 1];
        acc = wmma_f32(a, bb, acc);
    }

    const float bias   = b_ih[n0 + col] + b_hh[n0 + col]; // column-invariant per lane
    const int   rowoff = (lane >> 4) << 3;                // C rows: v + {0,8}
#pragma unroll
    for (int v = 0; v < 8; ++v) {
        const int row = m0 + v + rowoff;
        xg[(size_t)row * G4H_ + n0 + col] = acc[v] + bias;
    }
}

// ---------------------------------------------------------------------------
// Kernel 2: one LSTM timestep, fused gates + cell update.
// Block = 4 waves; wave w computes the 16x16 tile of gate w:
//   gates = xg_t + h_prev @ W_hh^T   (M=128, N=4H, K=512)
// then tiles meet in LDS for the elementwise cell update.
// grid=(8,32): 8*16=128 batch rows, 32*16=512 hidden cols.
// ---------------------------------------------------------------------------
__global__ __launch_bounds__(128) void lstm_step_kernel(
    const float* __restrict__ xg_t,    // [B,4H]  (xg + t*B*4H)
    const float* __restrict__ h_prev,  // [B,H]
    const float* __restrict__ c_prev,  // [B,H]
    const float* __restrict__ W_hh,    // [4H,H]
    float*       __restrict__ c_out,   // [B,H]
    float*       __restrict__ hs_t)    // [B,H]  (hs + t*B*H) == next h
{
    __shared__ float gtile[4][16][16];

    const int lane = threadIdx.x & 31;
    const int wave = threadIdx.x >> 5;     // gate index: 0=i 1=f 2=g 3=o
    const int col  = lane & 15;
    const int kb   = (lane >> 4) << 1;
    const int m0   = blockIdx.x * 16;      // batch tile
    const int j0   = blockIdx.y * 16;      // hidden tile
    const int n    = wave * H_ + j0 + col; // full gate column

    const float* arow = h_prev + (size_t)(m0 + col) * H_;
    const float* brow = W_hh  + (size_t)n * H_;          // B[k][n] = W_hh[n][k]

    v8f acc = {};
    for (int k = 0; k < H_; k += 4) {      // 128 iterations
        v2f a, bb;
        a.x  = arow[k + kb];  a.y  = arow[k + kb + 1];
        bb.x = brow[k + kb];  bb.y = brow[k + kb + 1];
        acc = wmma_f32(a, bb, acc);
    }

    const int rowoff = (lane >> 4) << 3;
#pragma unroll
    for (int v = 0; v < 8; ++v) {
        const int row = v + rowoff;        // local batch row 0..15
        gtile[wave][row][col] = acc[v] + xg_t[(size_t)(m0 + row) * G4H_ + n];
    }
    __syncthreads();

    // 256 tile elements across 128 threads
    for (int idx = threadIdx.x; idx < 256; idx += 128) {
        const int mm = idx >> 4;
        const int jj = idx & 15;
        const float ig = gtile[0][mm][jj];
        const float fg = gtile[1][mm][jj];
        const float gg = gtile[2][mm][jj];
        const float og = gtile[3][mm][jj];
        const size_t off = (size_t)(m0 + mm) * H_ + j0 + jj;
        const float cn = sigmoidf_(fg) * c_prev[off] + sigmoidf_(ig) * tanhf(gg);
        const float hn = sigmoidf_(og) * tanhf(cn);
        c_out[off] = cn;
        hs_t[off]  = hn;
    }
}

// ---------------------------------------------------------------------------
// Kernel 3: decode GEMM  out[b,t,v] = sum_h hs[t,b,h] * W_out[v,h] + b_out[v]
// M=T*B=2560, N=V=10000, K=512.
// Register blocking: each wave owns 2 M-tiles x 4 N-tiles (8 accumulators),
// so one k-step issues 8 WMMAs per 6 b64 loads (~10.7 FLOP/B from L1/L2).
// grid=(80,40): x covers 80*32=2560 rows; y covers 40*256 >= 10000 columns.
// ---------------------------------------------------------------------------
__global__ __launch_bounds__(128) void out_proj_kernel(
    const float* __restrict__ hs,      // [T*B, H], row r = t*B+b
    const float* __restrict__ W_out,   // [V,H]
    const float* __restrict__ b_out,   // [V]
    float*       __restrict__ out)     // [B,T,V]
{
    const int lane = threadIdx.x & 31;
    const int wave = threadIdx.x >> 5;
    const int col  = lane & 15;
    const int kb   = (lane >> 4) << 1;
    const int m0   = blockIdx.x * 32;                // two 16-row tiles
    const int nb   = blockIdx.y * 256 + wave * 64;   // first of 4 N-tiles

    int cnt = (V_ - nb + 15) >> 4;                   // # valid 16-wide tiles
    if (cnt <= 0) return;                            // wave-uniform exit
    if (cnt > 4) cnt = 4;

    const float* arow0 = hs + (size_t)(m0 + col) * H_;
    const float* arow1 = hs + (size_t)(m0 + 16 + col) * H_;

    v8f acc[2][4] = {};
    if (cnt == 4) {                                  // fast path: full 2x4 block
        for (int k = 0; k < H_; k += 4) {
            v2f a0, a1;
            a0.x = arow0[k + kb]; a0.y = arow0[k + kb + 1];
            a1.x = arow1[k + kb]; a1.y = arow1[k + kb + 1];
#pragma unroll
            for (int j = 0; j < 4; ++j) {
                const float* brow = W_out + (size_t)(nb + 16 * j + col) * H_;
                v2f bb;
                bb.x = brow[k + kb]; bb.y = brow[k + kb + 1];
                acc[0][j] = wmma_f32(a0, bb, acc[0][j]);
                acc[1][j] = wmma_f32(a1, bb, acc[1][j]);
            }
        }
    } else {                                         // tail: cnt in {1,2,3}
        for (int k = 0; k < H_; k += 4) {
            v2f a0, a1;
            a0.x = arow0[k + kb]; a0.y = arow0[k + kb + 1];
            a1.x = arow1[k + kb]; a1.y = arow1[k + kb + 1];
            for (int j = 0; j < cnt; ++j) {
                const float* brow = W_out + (size_t)(nb + 16 * j + col) * H_;
                v2f bb;
                bb.x = brow[k + kb]; bb.y = brow[k + kb + 1];
                acc[0][j] = wmma_f32(a0, bb, acc[0][j]);
                acc[1][j] = wmma_f32(a1, bb, acc[1][j]);
            }
        }
    }

    const int rowoff = (lane >> 4) << 3;
#pragma unroll
    for (int i = 0; i < 2; ++i) {
        for (int j = 0; j < cnt; ++j) {
            const int n = nb + 16 * j + col;
            const float bias = b_out[n];
#pragma unroll
            for (int v = 0; v < 8; ++v) {
                const int r = m0 + 16 * i + v + rowoff;  // r = t*B + b
                const int t = r / B_;
                const int b = r - t * B_;
                out[((size_t)b * T_ + t) * V_ + n] = acc[i][j][v] + bias;
            }
        }
    }
}

// ---------------------------------------------------------------------------
extern "C" void kernel_launch(void* const* d_in, const int* in_sizes, int n_in,
                              void* d_out, int out_size, void* d_ws, size_t ws_size,
                              hipStream_t stream) {
    const float* features = (const float*)d_in[0];   // [B,H]
    const int*   captions = (const int*)  d_in[1];   // [B,T]
    const float* embed    = (const float*)d_in[2];   // [V,E]
    const float* W_ih     = (const float*)d_in[3];   // [4H,E]
    const float* W_hh     = (const float*)d_in[4];   // [4H,H]
    const float* b_ih     = (const float*)d_in[5];   // [4H]
    const float* b_hh     = (const float*)d_in[6];   // [4H]
    const float* W_out    = (const float*)d_in[7];   // [V,H]
    const float* b_out    = (const float*)d_in[8];   // [V]
    float* out = (float*)d_out;                      // [B,T,V]

    // Workspace layout (floats): xg[T*B*4H] | hs[T*B*H] | cbuf0[B*H] | cbuf1[B*H]
    float* xg    = (float*)d_ws;
    float* hs    = xg + (size_t)T_ * B_ * G4H_;
    float* cbuf0 = hs + (size_t)T_ * B_ * H_;
    float* cbuf1 = cbuf0 + (size_t)B_ * H_;

    // c0 = 0 (cbuf1 is c_prev at t=0); memset node is graph-capture legal
    hipMemsetAsync(cbuf1, 0, (size_t)B_ * H_ * sizeof(float), stream);

    // Phase 1: all-timestep input projection (parallel over T)
    xg_kernel<<<dim3(160, 32), 128, 0, stream>>>(captions, embed, W_ih, b_ih, b_hh, xg);

    // Phase 2: sequential recurrence; h chains through hs, c ping-pongs
    for (int t = 0; t < T_; ++t) {
        const float* h_prev = (t == 0) ? features : hs + (size_t)(t - 1) * B_ * H_;
        const float* c_prev = (t & 1) ? cbuf0 : cbuf1;
        float*       c_next = (t & 1) ? cbuf1 : cbuf0;
        lstm_step_kernel<<<dim3(8, 32), 128, 0, stream>>>(
            xg + (size_t)t * B_ * G4H_, h_prev, c_prev, W_hh,
            c_next, hs + (size_t)t * B_ * H_);
    }

    // Phase 3: decode to vocab logits
    out_proj_kernel<<<dim3(80, 40), 128, 0, stream>>>(hs, W_out, b_out, out);
}